// VectorQuantizer_60043642798098
// MI455X (gfx1250) — compile-verified
//
#include <hip/hip_runtime.h>
#include <hip/hip_bf16.h>
#include <math.h>

// Problem constants (B,H,W,D,K) = (32,64,64,64,512)
static constexpr int Np   = 131072;       // B*H*W
static constexpr int Dp   = 64;
static constexpr int Kp   = 512;

// d_out flat layout (tuple concatenated, element offsets)
static constexpr size_t QOFF  = 0;                    // quantized_st: N*D
static constexpr size_t LOSSO = 8388608;              // loss: 1
static constexpr size_t PERPO = 8388609;              // perplexity: 1
static constexpr size_t ENCO  = 8388610;              // encodings: N*K
static constexpr size_t IDXO  = 75497474;             // encoding_indices: N
static constexpr size_t DISTO = 75628546;             // distances: N*K

typedef float v2f __attribute__((ext_vector_type(2)));
typedef float v8f __attribute__((ext_vector_type(8)));

// ---------------------------------------------------------------------------
// Kernel 0: codebook squared norms ||e_k||^2 into ws[0..511]; zero accumulators
// ---------------------------------------------------------------------------
__global__ void vq_enorm_kernel(const float* __restrict__ E, float* __restrict__ ws) {
    int k = threadIdx.x;                      // 512 threads
    float s = 0.f;
    #pragma unroll 8
    for (int d = 0; d < Dp; ++d) {
        float e = E[(size_t)d * Kp + k];
        s += e * e;
    }
    ws[k] = s;
    if (k < 2) ws[Kp + k] = 0.f;              // ws[512]=loss acc, ws[513]=entropy acc
}

// ---------------------------------------------------------------------------
// Main fused kernel: one wave32 per 16-row strip.
//   distances tile via V_WMMA_F32_16X16X4_F32 (16 K-steps over D=64),
//   inline writes of distances + zeroed encodings, running argmin,
//   then codebook gather -> quantized_st, one-hot, indices, loss partial.
// ---------------------------------------------------------------------------
#define WAVES_PER_BLOCK 4
__global__ __launch_bounds__(32 * WAVES_PER_BLOCK)
void vq_main_kernel(const float* __restrict__ x,   // (N, 64)
                    const float* __restrict__ E,   // (64, 512)
                    const float* __restrict__ enorm,
                    float* __restrict__ out,
                    float* __restrict__ lossAcc,
                    int*   __restrict__ idxWs) {
    __shared__ int s_idx[WAVES_PER_BLOCK][16];

    const int lane  = threadIdx.x & 31;
    const int wave  = threadIdx.x >> 5;
    const int strip = blockIdx.x * WAVES_PER_BLOCK + wave;   // 0..8191
    const int row0  = strip * 16;
    const int ln    = lane & 15;     // column-within-tile / row-within-A
    const int hh    = lane >> 4;     // lane half

    // ---- preload A tile (16x64 fp32) into 16 float2 regs per lane ----
    // A layout (32-bit 16x4): lane L holds row L&15; VGPR0/1 = K{0,1} (half 0)
    // or K{2,3} (half 1) of each 4-wide K step.
    v2f a[16];
    const float* xrow = x + (size_t)(row0 + ln) * Dp + 2 * hh;
    float xn = 0.f;
    #pragma unroll
    for (int s = 0; s < 16; ++s) {
        a[s] = *(const v2f*)(xrow + 4 * s);
        xn += a[s].x * a[s].x + a[s].y * a[s].y;
    }
    // combine lane halves: lanes L and L^16 together hold full row L&15
    xn += __shfl_xor(xn, 16, 32);
    // broadcast per-C-row norms: C VGPR j holds row j (half 0) / j+8 (half 1)
    float xnj[8];
    #pragma unroll
    for (int j = 0; j < 8; ++j) xnj[j] = __shfl(xn, j + 8 * hh, 32);

    float minv[8];
    int   mini[8];
    #pragma unroll
    for (int j = 0; j < 8; ++j) { minv[j] = 3.4e38f; mini[j] = 0; }

    float* distBase = out + DISTO + (size_t)row0 * Kp;
    float* encBase  = out + ENCO  + (size_t)row0 * Kp;

    // ---- sweep 32 column tiles of the codebook ----
    for (int n0 = 0; n0 < Kp; n0 += 16) {
        const int n = n0 + ln;
        const float en = enorm[n];
        const float* Eb = E + (size_t)(2 * hh) * Kp + n;

        v8f c = {0.f, 0.f, 0.f, 0.f, 0.f, 0.f, 0.f, 0.f};
        #pragma unroll
        for (int s = 0; s < 16; ++s) {
            v2f b;
            b.x = Eb[(size_t)(4 * s)     * Kp];
            b.y = Eb[(size_t)(4 * s + 1) * Kp];
            // D = A(16x4) * B(4x16) + C   (fp32 WMMA, 8-arg form)
            c = __builtin_amdgcn_wmma_f32_16x16x4_f32(
                    false, a[s], false, b, (short)0, c, false, false);
        }

        #pragma unroll
        for (int j = 0; j < 8; ++j) {
            const int m = j + 8 * hh;            // row within strip
            float dist = xnj[j] + en - 2.0f * c[j];
            distBase[(size_t)m * Kp + n] = dist;
            encBase [(size_t)m * Kp + n] = 0.0f; // one-hot zero fill
            if (dist < minv[j] || (dist == minv[j] && n < mini[j])) {
                minv[j] = dist; mini[j] = n;
            }
        }
    }

    // ---- argmin reduce across the 16 lanes of each half (prefer low index) ----
    #pragma unroll
    for (int j = 0; j < 8; ++j) {
        #pragma unroll
        for (int m2 = 8; m2 >= 1; m2 >>= 1) {
            float ov = __shfl_xor(minv[j], m2, 32);
            int   oi = __shfl_xor(mini[j], m2, 32);
            if (ov < minv[j] || (ov == minv[j] && oi < mini[j])) {
                minv[j] = ov; mini[j] = oi;
            }
        }
    }

    if (ln == 0) {   // lanes 0 and 16 own rows j and j+8 respectively
        #pragma unroll
        for (int j = 0; j < 8; ++j) {
            const int m  = j + 8 * hh;
            const int id = mini[j];
            s_idx[wave][m] = id;
            encBase[(size_t)m * Kp + id] = 1.0f;         // one-hot set
            out[IDXO + row0 + m] = (float)id;            // encoding_indices
            idxWs[row0 + m] = id;                        // int copy for perplexity
        }
    }
    __syncthreads();

    // ---- codebook gather -> quantized_st (== quantized in fwd) + loss partial ----
    float acc = 0.f;
    const int d0 = lane * 2;
    #pragma unroll 4
    for (int m = 0; m < 16; ++m) {
        const int id = s_idx[wave][m];
        float q0 = E[(size_t)d0       * Kp + id];
        float q1 = E[(size_t)(d0 + 1) * Kp + id];
        const float* xr = x + (size_t)(row0 + m) * Dp + d0;
        float x0 = xr[0], x1 = xr[1];
        v2f q = {q0, q1};
        *(v2f*)(out + QOFF + (size_t)(row0 + m) * Dp + d0) = q;
        float e0 = q0 - x0, e1 = q1 - x1;
        acc += e0 * e0 + e1 * e1;
    }
    #pragma unroll
    for (int m2 = 16; m2 >= 1; m2 >>= 1) acc += __shfl_xor(acc, m2, 32);
    if (lane == 0) atomicAdd(lossAcc, acc);
}

// ---------------------------------------------------------------------------
// Perplexity: avg over batch axis only -> one thread per (h,w), 32 indices.
// ---------------------------------------------------------------------------
__global__ void vq_perp_kernel(const int* __restrict__ idxWs, float* __restrict__ entAcc) {
    const int t = blockIdx.x * blockDim.x + threadIdx.x;   // 0..4095 (h*64+w)
    int ids[32];
    #pragma unroll
    for (int b = 0; b < 32; ++b) ids[b] = idxWs[b * 4096 + t];
    float S = 0.f;
    for (int i = 0; i < 32; ++i) {
        bool first = true;
        for (int j = 0; j < i; ++j)
            if (ids[j] == ids[i]) { first = false; break; }
        if (first) {
            int c = 1;
            for (int j = i + 1; j < 32; ++j) c += (ids[j] == ids[i]);
            float p = (float)c * (1.0f / 32.0f);
            S += p * logf(p + 1e-10f);
        }
    }
    #pragma unroll
    for (int m2 = 16; m2 >= 1; m2 >>= 1) S += __shfl_xor(S, m2, 32);
    if ((threadIdx.x & 31) == 0) atomicAdd(entAcc, S);
}

// ---------------------------------------------------------------------------
// Finalize scalars.
// ---------------------------------------------------------------------------
__global__ void vq_final_kernel(const float* __restrict__ ws, float* __restrict__ out) {
    if (threadIdx.x == 0) {
        // loss = q_latent + 0.25*e_latent ; both equal mean((q-x)^2) numerically
        out[LOSSO] = 1.25f * ws[Kp] * (1.0f / 8388608.0f);
        out[PERPO] = expf(-ws[Kp + 1]);
    }
}

extern "C" void kernel_launch(void* const* d_in, const int* in_sizes, int n_in,
                              void* d_out, int out_size, void* d_ws, size_t ws_size,
                              hipStream_t stream) {
    const float* x = (const float*)d_in[0];   // (32,64,64,64)
    const float* E = (const float*)d_in[1];   // (64,512)
    float* out = (float*)d_out;

    float* wsf   = (float*)d_ws;              // [0..511] enorm, [512] loss, [513] entropy
    int*   idxWs = (int*)(wsf + Kp + 2);      // 131072 ints

    vq_enorm_kernel<<<1, Kp, 0, stream>>>(E, wsf);
    vq_main_kernel<<<(Np / 16) / WAVES_PER_BLOCK, 32 * WAVES_PER_BLOCK, 0, stream>>>(
        x, E, wsf, out, wsf + Kp, idxWs);
    vq_perp_kernel<<<16, 256, 0, stream>>>(idxWs, wsf + Kp + 1);
    vq_final_kernel<<<1, 64, 0, stream>>>(wsf, out);
}